// NeuralNetwork_23733989277834
// MI455X (gfx1250) — compile-verified
//
#include <hip/hip_runtime.h>

// ---------------------------------------------------------------------------
// Problem constants (match reference): B=4096, T=512, D=18, H=8, O=1, 4H=32.
// Kernel 1: xg[B*T, 32] = x[B*T,18] @ W_ih^T + (b_ih+b_hh)   via v_wmma f32.
// Kernel 2: one wave32 per sequence; lane = gate; LSTM recurrence + O=1 head.
// ---------------------------------------------------------------------------

typedef float v2f __attribute__((ext_vector_type(2)));
typedef float v8f __attribute__((ext_vector_type(8)));

#define B_ 4096
#define T_ 512
#define D_ 18
#define G_ 32   // 4*H
#define TPW 8   // output tiles (16 rows each) per wave in kernel 1

// ---------------------------------------------------------------------------
// Kernel 1: input projection GEMM with V_WMMA_F32_16X16X4_F32.
// M = B*T rows, K = 18 (padded to 20 = 5 k-blocks of 4), N = 32 (2 tiles).
// A 16x4 layout: lanes 0-15 -> K={0,1}, lanes 16-31 -> K={2,3} (2 VGPRs).
// B 4x16 layout: lane%16 = column, same K split (2 VGPRs). C: 8 VGPRs.
// ---------------------------------------------------------------------------
__global__ __launch_bounds__(256) void xgemm_wmma(
    const float* __restrict__ x, const float* __restrict__ Wih,
    const float* __restrict__ bih, const float* __restrict__ bhh,
    float* __restrict__ xg)
{
  const int lane = threadIdx.x & 31;
  const int wave = threadIdx.x >> 5;
  const int m    = lane & 15;
  const bool hi  = lane >= 16;

  // Preload B (W_ih^T in WMMA B layout, zero-padded K=18..19) + bias, reused
  // across TPW tiles.
  v2f breg[5][2];
  float bias[2];
#pragma unroll
  for (int nt = 0; nt < 2; ++nt) {
    const int col = nt * 16 + m;            // gate index 0..31 (torch i,f,g,o)
    bias[nt] = bih[col] + bhh[col];
#pragma unroll
    for (int kb = 0; kb < 5; ++kb) {
      const int k0 = kb * 4 + (hi ? 2 : 0);
      v2f v;
      v.x = (k0     < D_) ? Wih[col * D_ + k0]     : 0.0f;
      v.y = (k0 + 1 < D_) ? Wih[col * D_ + k0 + 1] : 0.0f;
      breg[kb][nt] = v;
    }
  }

  long row0 = ((long)blockIdx.x * 8 + wave) * (16 * TPW);
  for (int tile = 0; tile < TPW; ++tile, row0 += 16) {
    const long row = row0 + m;
    v2f areg[5];
#pragma unroll
    for (int kb = 0; kb < 5; ++kb) {
      const int k0 = kb * 4 + (hi ? 2 : 0);
      v2f v = {0.0f, 0.0f};
      if (k0 < D_)  // k0 is even, <=16 -> 8B-aligned float2, k0+1<=17 valid
        v = *(const v2f*)(x + row * D_ + k0);
      areg[kb] = v;
    }
    v8f c0 = {};
    v8f c1 = {};
#pragma unroll
    for (int kb = 0; kb < 5; ++kb) {
      c0 = __builtin_amdgcn_wmma_f32_16x16x4_f32(false, areg[kb], false,
                                                 breg[kb][0], (short)0, c0,
                                                 false, false);
      c1 = __builtin_amdgcn_wmma_f32_16x16x4_f32(false, areg[kb], false,
                                                 breg[kb][1], (short)0, c1,
                                                 false, false);
    }
    // C layout: VGPR e holds M=e (lanes 0-15) and M=e+8 (lanes 16-31), N=m.
    const long rbase = row0 + (hi ? 8 : 0);
#pragma unroll
    for (int e = 0; e < 8; ++e) {
      const long r = rbase + e;
      xg[r * G_ +      m] = c0[e] + bias[0];
      xg[r * G_ + 16 + m] = c1[e] + bias[1];
    }
  }
}

// ---------------------------------------------------------------------------
// Kernel 2: LSTM recurrence + linear head. One wave per batch sequence.
// Lane g in [0,32) owns gate g (order: i[0:8], f[8:16], g[16:24], o[24:32]).
// h/c state replicated as h[lane&7] in every lane (bpermute gathers do this
// for free). Recurrent matvec via 8 x (v_readlane + v_fma). Branch on t<len
// is wave-uniform (one sequence per wave) -> cheap scalar skip for padding.
// ---------------------------------------------------------------------------
__global__ __launch_bounds__(256) void lstm_head(
    const float* __restrict__ xg, const int* __restrict__ lengths,
    const float* __restrict__ Whh, const float* __restrict__ Wlin,
    const float* __restrict__ blin, float* __restrict__ out)
{
  const int lane = threadIdx.x & 31;
  const int wave = threadIdx.x >> 5;
  const int b    = blockIdx.x * 8 + wave;
  const int len  = lengths[b];          // wave-uniform
  const int hh8  = lane & 7;

  // W_hh column strip for this lane's gate: W_hh[lane, 0..7]
  float wcol[8];
#pragma unroll
  for (int k = 0; k < 8; ++k) wcol[k] = Whh[lane * 8 + k];

  // Unified activation: act(x) = A * rcp(1 + exp2(Bc*x)) + C
  // sigmoid: A=1, Bc=-log2e,  C=0   (lanes 0-15, 24-31: i, f, o)
  // tanh:    A=2, Bc=-2log2e, C=-1  (lanes 16-23: g)
  const float LOG2E = 1.4426950408889634f;
  const bool isTanh = (lane >= 16) && (lane < 24);
  const float Bc = isTanh ? (-2.0f * LOG2E) : (-LOG2E);
  const float Ac = isTanh ? 2.0f : 1.0f;
  const float Cc = isTanh ? -1.0f : 0.0f;

  const float wl = Wlin[hh8];   // O = 1
  const float bl = blin[0];

  float h = 0.0f, c = 0.0f;
  float ybuf = 0.0f;
  const float* xgp = xg + (size_t)b * T_ * G_ + lane;
  float* yp = out + (size_t)b * T_;

  for (int t = 0; t < T_; ++t) {
    float yv = 0.0f;
    if (t < len) {                       // uniform branch: padding is free
      float gacc = xgp[(size_t)t * G_];  // coalesced 128B per wave
      // stream-ahead prefetch of future xg rows (global_prefetch_b8)
      int tp = t + 24; tp = tp < T_ ? tp : (T_ - 1);
      __builtin_prefetch(xgp + (size_t)tp * G_, 0, 1);

#pragma unroll
      for (int k = 0; k < 8; ++k) {      // gates += h @ W_hh^T
        const float hk = __uint_as_float(
            __builtin_amdgcn_readlane(__float_as_uint(h), k));
        gacc = __builtin_fmaf(wcol[k], hk, gacc);
      }
      const float e   = __builtin_amdgcn_exp2f(gacc * Bc);
      const float r   = __builtin_amdgcn_rcpf(1.0f + e);
      const float act = __builtin_fmaf(Ac, r, Cc);

      // gather i,f,g,o for unit hh8 into every lane (ds_bpermute)
      const float iv = __shfl(act, hh8,      32);
      const float fv = __shfl(act, hh8 + 8,  32);
      const float gv = __shfl(act, hh8 + 16, 32);
      const float ov = __shfl(act, hh8 + 24, 32);

      c = __builtin_fmaf(fv, c, iv * gv);
      const float e2 = __builtin_amdgcn_exp2f(c * (-2.0f * LOG2E));
      const float tc = __builtin_fmaf(2.0f, __builtin_amdgcn_rcpf(1.0f + e2),
                                      -1.0f);
      h = ov * tc;                       // h[lane&7], replicated in all lanes

      // O=1 head: y = sum_hh W_lin[hh]*h[hh] + b (xor-reduce over 8-group)
      float p = h * wl;
      p += __shfl_xor(p, 1, 32);
      p += __shfl_xor(p, 2, 32);
      p += __shfl_xor(p, 4, 32);
      yv = p + bl;
    }
    // stage y in lane (t&31); flush 32 timesteps with one coalesced store
    ybuf = (lane == (t & 31)) ? yv : ybuf;
    if ((t & 31) == 31)
      yp[(t & ~31) + lane] = ybuf;       // T_ % 32 == 0 -> no tail
  }
}

// ---------------------------------------------------------------------------
extern "C" void kernel_launch(void* const* d_in, const int* in_sizes, int n_in,
                              void* d_out, int out_size, void* d_ws,
                              size_t ws_size, hipStream_t stream) {
  (void)in_sizes; (void)n_in; (void)out_size; (void)ws_size;
  const float* x    = (const float*)d_in[0];
  const int*   lens = (const int*)d_in[1];
  const float* Wih  = (const float*)d_in[2];
  const float* Whh  = (const float*)d_in[3];
  const float* bih  = (const float*)d_in[4];
  const float* bhh  = (const float*)d_in[5];
  const float* Wlin = (const float*)d_in[6];
  const float* blin = (const float*)d_in[7];
  float* out = (float*)d_out;
  float* xg  = (float*)d_ws;   // B*T*32 floats = 256 MiB scratch

  const long R      = (long)B_ * T_;            // 2,097,152 rows
  const long tiles  = R / 16;                   // 131,072
  const int blocks1 = (int)(tiles / (TPW * 8)); // 8 waves/block
  hipLaunchKernelGGL(xgemm_wmma, dim3(blocks1), dim3(256), 0, stream,
                     x, Wih, bih, bhh, xg);
  hipLaunchKernelGGL(lstm_head, dim3(B_ / 8), dim3(256), 0, stream,
                     xg, lens, Whh, Wlin, blin, out);
}